// ExpertAllocation_70214125355035
// MI455X (gfx1250) — compile-verified
//
#include <hip/hip_runtime.h>

#define D_DIM 4096
#define E_NUM 64
#define TOKS  16384
#define RBLKS 256      // router blocks, 64 tokens each (4 waves x 16 tokens)
#define CAPF  256.0    // int(T/E * capacity_factor)

typedef float v2f __attribute__((ext_vector_type(2)));
typedef float v8f __attribute__((ext_vector_type(8)));

// ---------------------------------------------------------------------------
// Repack W (D x E row-major) so that a WMMA B-fragment lane load is one b64:
// Wp[q][n] = { W[2q][n], W[2q+1][n] } for q in [0, D/2)
// ---------------------------------------------------------------------------
__global__ __launch_bounds__(256) void pack_w_kernel(const float* __restrict__ W,
                                                     float* __restrict__ Wp) {
  int i = blockIdx.x * 256 + threadIdx.x;          // 0 .. D/2*E - 1
  if (i >= (D_DIM / 2) * E_NUM) return;
  int q = i >> 6, n = i & 63;
  v2f v;
  v.x = W[(2 * q) * E_NUM + n];
  v.y = W[(2 * q + 1) * E_NUM + n];
  ((v2f*)Wp)[i] = v;
}

// ---------------------------------------------------------------------------
// Router: WMMA fp32 GEMM (16 tokens/wave x 64 experts) + softmax + top-1.
// Writes probs into out[t*64+e], top idx/prob, and per-block partials.
// ---------------------------------------------------------------------------
__global__ __launch_bounds__(128) void router_kernel(
    const float* __restrict__ x, const float* __restrict__ Wp,
    const float* __restrict__ bias, float* __restrict__ out,
    int* __restrict__ top_idx, float* __restrict__ top_prob,
    float* __restrict__ Pblk, float* __restrict__ fblk, int* __restrict__ cblk) {
  __shared__ float lds[4][16][E_NUM + 1];  // logits -> probs
  __shared__ int   amax[64];
  __shared__ float aprob[64];

  const int tid  = threadIdx.x;
  const int wave = tid >> 5;
  const int lane = tid & 31;
  const int m    = lane & 15;   // token row within tile / N column
  const int h    = lane >> 4;   // lane half (K split)
  const int tok0 = (blockIdx.x * 4 + wave) * 16;

  const float* xrow = x + (size_t)(tok0 + m) * D_DIM;

  v8f c0{}, c1{}, c2{}, c3{};
  for (int k0 = 0; k0 < D_DIM; k0 += 4) {
    // A 16x4 f32 fragment: VGPR0=K(2h), VGPR1=K(2h+1) in this lane half
    v2f a = *(const v2f*)(xrow + k0 + 2 * h);
    // B 4x16 f32 fragments for the four N tiles (packed pairs of K rows)
    const v2f* wq = (const v2f*)Wp + (size_t)(k0 / 2 + h) * E_NUM + m;
    v2f b0 = wq[0];
    v2f b1 = wq[16];
    v2f b2 = wq[32];
    v2f b3 = wq[48];
    c0 = __builtin_amdgcn_wmma_f32_16x16x4_f32(false, a, false, b0, (short)0, c0, false, false);
    c1 = __builtin_amdgcn_wmma_f32_16x16x4_f32(false, a, false, b1, (short)0, c1, false, false);
    c2 = __builtin_amdgcn_wmma_f32_16x16x4_f32(false, a, false, b2, (short)0, c2, false, false);
    c3 = __builtin_amdgcn_wmma_f32_16x16x4_f32(false, a, false, b3, (short)0, c3, false, false);
  }

  // C layout: VGPR r holds row (r + 8*h), column 16*t + m
#pragma unroll
  for (int r = 0; r < 8; ++r) {
    int row = r + 8 * h;
    lds[wave][row][m]      = c0[r];
    lds[wave][row][16 + m] = c1[r];
    lds[wave][row][32 + m] = c2[r];
    lds[wave][row][48 + m] = c3[r];
  }
  __syncthreads();

  // Softmax + argmax: threads 0..63 each own one of the block's 64 tokens
  if (tid < 64) {
    const int w = tid >> 4, row = tid & 15;
    const int token = blockIdx.x * 64 + tid;
    float mx = -3.4e38f;
#pragma unroll 8
    for (int e = 0; e < E_NUM; ++e) {
      float l = lds[w][row][e] + bias[e];
      lds[w][row][e] = l;
      mx = fmaxf(mx, l);
    }
    float s = 0.f;
#pragma unroll 8
    for (int e = 0; e < E_NUM; ++e) s += expf(lds[w][row][e] - mx);
    float inv = 1.f / s;
    int   best  = 0;
    float bestp = -1.f;
    float* orow = out + (size_t)token * E_NUM;
#pragma unroll 8
    for (int e = 0; e < E_NUM; ++e) {
      float p = expf(lds[w][row][e] - mx) * inv;
      lds[w][row][e] = p;
      orow[e] = p;
      if (p > bestp) { bestp = p; best = e; }  // first-max tie-break
    }
    top_idx[token]  = best;
    top_prob[token] = bestp;
    amax[tid]  = best;
    aprob[tid] = bestp;
  }
  __syncthreads();

  // Per-block deterministic partials for P_i, f_i, counts
  if (tid < 64) {
    const int e = tid;
    float Ps = 0.f, fs = 0.f;
    int cnt = 0;
#pragma unroll 4
    for (int i = 0; i < 64; ++i) {
      Ps += lds[i >> 4][i & 15][e];
      if (amax[i] == e) { fs += aprob[i]; ++cnt; }
    }
    Pblk[blockIdx.x * 64 + e] = Ps;
    fblk[blockIdx.x * 64 + e] = fs;
    cblk[blockIdx.x * 64 + e] = cnt;
  }
}

// ---------------------------------------------------------------------------
// Finalize: reduce per-block partials, prefix-sum expert counts, aux loss.
// ---------------------------------------------------------------------------
__global__ __launch_bounds__(64) void finalize_kernel(
    const float* __restrict__ Pblk, const float* __restrict__ fblk,
    const int* __restrict__ cblk, int* __restrict__ count,
    int* __restrict__ offset, int* __restrict__ cursor,
    float* __restrict__ out) {
  __shared__ float term[64];
  const int e = threadIdx.x;
  double P = 0.0, f = 0.0;
  int c = 0;
  for (int b = 0; b < RBLKS; ++b) {
    P += Pblk[b * 64 + e];
    f += fblk[b * 64 + e];
    c += cblk[b * 64 + e];
  }
  count[e] = c;
  const double T = (double)TOKS;
  term[e] = (float)((f / T) * (P / T));
  __syncthreads();
  if (e == 0) {
    int off = 0;
    float aux = 0.f;
    for (int i = 0; i < 64; ++i) {
      offset[i] = off;
      cursor[i] = off;
      off += count[i];
      aux += term[i];
    }
    out[(size_t)TOKS * E_NUM] = 0.01f * (float)E_NUM * aux;  // ALPHA * E * sum
  }
}

// ---------------------------------------------------------------------------
// Scatter tokens into per-expert segments (counting sort; order irrelevant).
// ---------------------------------------------------------------------------
__global__ __launch_bounds__(256) void scatter_kernel(
    const int* __restrict__ top_idx, const float* __restrict__ top_prob,
    int* __restrict__ cursor, int* __restrict__ list_tok,
    float* __restrict__ list_prob) {
  int t = blockIdx.x * 256 + threadIdx.x;
  if (t >= TOKS) return;
  int e = top_idx[t];
  int slot = atomicAdd(&cursor[e], 1);
  list_tok[slot]  = t;
  list_prob[slot] = top_prob[t];
}

// ---------------------------------------------------------------------------
// Capacity assignment: token kept iff sum of probs ranked ahead of it
// (prob desc, token-index asc tie-break) plus its own prob <= capacity.
// Equivalent to the reference's sorted-cumsum <= capacity mask.
// ---------------------------------------------------------------------------
__global__ __launch_bounds__(256) void assign_kernel(
    const int* __restrict__ top_idx, const float* __restrict__ top_prob,
    const int* __restrict__ offset, const int* __restrict__ count,
    const int* __restrict__ list_tok, const float* __restrict__ list_prob,
    float* __restrict__ out) {
  int t = blockIdx.x * 256 + threadIdx.x;
  if (t >= TOKS) return;
  const int   e = top_idx[t];
  const float p = top_prob[t];
  const int o = offset[e], c = count[e];
  double s = 0.0;
  for (int j = o; j < o + c; ++j) {
    float pj = list_prob[j];
    int   tj = list_tok[j];
    if (pj > p || (pj == p && tj <= t)) s += (double)pj;
  }
  if (s > CAPF) {
    float* orow = out + (size_t)t * E_NUM;
#pragma unroll 8
    for (int e2 = 0; e2 < E_NUM; ++e2) orow[e2] = 0.f;
  }
}

// ---------------------------------------------------------------------------
extern "C" void kernel_launch(void* const* d_in, const int* in_sizes, int n_in,
                              void* d_out, int out_size, void* d_ws, size_t ws_size,
                              hipStream_t stream) {
  (void)in_sizes; (void)n_in; (void)out_size; (void)ws_size;
  const float* x = (const float*)d_in[0];   // (4,4096,4096)
  const float* W = (const float*)d_in[1];   // (4096,64)
  const float* b = (const float*)d_in[2];   // (64,)
  float* out = (float*)d_out;               // T*64 probs + 1 aux scalar

  char* ws = (char*)d_ws;
  size_t off = 0;
  auto alloc = [&](size_t bytes) { char* p = ws + off; off = (off + bytes + 255) & ~(size_t)255; return p; };

  float* Wp        = (float*)alloc((size_t)(D_DIM / 2) * E_NUM * 2 * sizeof(float)); // 1 MB
  int*   top_idx   = (int*)  alloc(TOKS * sizeof(int));
  float* top_prob  = (float*)alloc(TOKS * sizeof(float));
  int*   list_tok  = (int*)  alloc(TOKS * sizeof(int));
  float* list_prob = (float*)alloc(TOKS * sizeof(float));
  float* Pblk      = (float*)alloc(RBLKS * 64 * sizeof(float));
  float* fblk      = (float*)alloc(RBLKS * 64 * sizeof(float));
  int*   cblk      = (int*)  alloc(RBLKS * 64 * sizeof(int));
  int*   count     = (int*)  alloc(64 * sizeof(int));
  int*   offset    = (int*)  alloc(64 * sizeof(int));
  int*   cursor    = (int*)  alloc(64 * sizeof(int));

  pack_w_kernel<<<((D_DIM / 2) * E_NUM + 255) / 256, 256, 0, stream>>>(W, Wp);
  router_kernel<<<RBLKS, 128, 0, stream>>>(x, Wp, b, out, top_idx, top_prob,
                                           Pblk, fblk, cblk);
  finalize_kernel<<<1, 64, 0, stream>>>(Pblk, fblk, cblk, count, offset, cursor, out);
  scatter_kernel<<<TOKS / 256, 256, 0, stream>>>(top_idx, top_prob, cursor,
                                                 list_tok, list_prob);
  assign_kernel<<<TOKS / 256, 256, 0, stream>>>(top_idx, top_prob, offset, count,
                                                list_tok, list_prob, out);
}